// AttentionBlockE3_23424751632788
// MI455X (gfx1250) — compile-verified
//
#include <hip/hip_runtime.h>

// ---------------------------------------------------------------------------
// E(3)-equivariant multi-head graph attention for MI455X (gfx1250, wave32).
//
// Pipeline (all on `stream`, graph-capture safe, deterministic):
//   1. zero_i32      : clear count/rowptr/cursor scratch
//   2. count_deg     : degree histogram over dst nodes (int atomics, tiny)
//   3. scan_rowptr   : single-block exclusive scan -> CSR row pointers
//   4. scatter_perm  : bucket edge ids per dst node
//   5. logits_wmma   : W[e,h] = ewc[e]/sqrt(60) * <q_h, k_h>  via
//                      V_WMMA_F32_16X16X4_F32, W16x8 = (Q.*K)16x480 x Sel480x8
//                      with the head-selection B built as a branchless
//                      running-interval test per irrep region
//   6. aggregate     : per-node online softmax + weighted sum of v
//                      (atomic-free gather, deterministic order via LDS sort)
// ---------------------------------------------------------------------------

typedef __attribute__((ext_vector_type(2))) float v2f;
typedef __attribute__((ext_vector_type(8))) float v8f;

#define NHEADS     8
#define DFUSE      480
#define INV_SQRT60 0.1290994448735805628f

// fused-column -> head, from irrep blocks [128 | 192 | 160],
// per-head chunks of [16 | 24 | 20].  (Used only outside hot loops.)
__device__ __forceinline__ int head_of_col(int c) {
    if (c < 128) return c >> 4;
    if (c < 320) return (c - 128) / 24;
    return (c - 320) / 20;
}

// ---------------------------------------------------------------------------
__global__ void zero_i32(int* __restrict__ p, int n) {
    int i = blockIdx.x * blockDim.x + threadIdx.x;
    if (i < n) p[i] = 0;
}

__global__ void count_deg(const long long* __restrict__ dst,
                          int* __restrict__ cnt, int E) {
    int e = blockIdx.x * blockDim.x + threadIdx.x;
    if (e < E) atomicAdd(&cnt[(int)dst[e]], 1);
}

// single 1024-thread block: exclusive prefix sum of cnt[0..N) -> rowptr[0..N]
__global__ void scan_rowptr(const int* __restrict__ cnt,
                            int* __restrict__ rowptr, int N) {
    __shared__ int sh[1024];
    __shared__ int carry;
    const int t = threadIdx.x;
    if (t == 0) carry = 0;
    __syncthreads();
    for (int base = 0; base < N; base += 1024) {
        const int i = base + t;
        sh[t] = (i < N) ? cnt[i] : 0;
        __syncthreads();
        for (int off = 1; off < 1024; off <<= 1) {
            int add = (t >= off) ? sh[t - off] : 0;
            __syncthreads();
            sh[t] += add;
            __syncthreads();
        }
        if (i < N) rowptr[i + 1] = carry + sh[t];
        __syncthreads();
        if (t == 0) carry += sh[1023];
        __syncthreads();
    }
    if (t == 0) rowptr[0] = 0;
}

__global__ void scatter_perm(const long long* __restrict__ dst,
                             const int* __restrict__ rowptr,
                             int* __restrict__ cursor,
                             int* __restrict__ perm, int E) {
    int e = blockIdx.x * blockDim.x + threadIdx.x;
    if (e < E) {
        int n = (int)dst[e];
        int pos = rowptr[n] + atomicAdd(&cursor[n], 1);
        perm[pos] = e;
    }
}

// ---------------------------------------------------------------------------
// One irrep region of the logits WMMA loop.
//   qp/kp point at  row + region_start + 2*half  (this lane's first column)
//   t    = c0 - (region_start + m*SPAN) : in-head-interval counter, step 4
//   b    = 1.0 iff current column-pair belongs to head m (pair never straddles
//          a head boundary: spans & starts are even, c0 is even)
// Two accumulators break the WMMA RAW dependency chain (nchunks is even).
// ---------------------------------------------------------------------------
template <int SPAN, int NCHUNKS>
__device__ __forceinline__ void
logits_region(const float* __restrict__ qp, const float* __restrict__ kp,
              int t, v8f& acc0, v8f& acc1) {
#pragma unroll 4
    for (int i = 0; i < NCHUNKS; i += 2) {
        v2f q0 = *(const v2f*)(qp + 4 * i);
        v2f k0 = *(const v2f*)(kp + 4 * i);
        v2f q1 = *(const v2f*)(qp + 4 * i + 4);
        v2f k1 = *(const v2f*)(kp + 4 * i + 4);
        v2f a0, a1, b0, b1;
        a0.x = q0.x * k0.x; a0.y = q0.y * k0.y;
        a1.x = q1.x * k1.x; a1.y = q1.y * k1.y;
        float s0 = ((unsigned)t       < (unsigned)SPAN) ? 1.0f : 0.0f;
        float s1 = ((unsigned)(t + 4) < (unsigned)SPAN) ? 1.0f : 0.0f;
        b0.x = s0; b0.y = s0;
        b1.x = s1; b1.y = s1;
        acc0 = __builtin_amdgcn_wmma_f32_16x16x4_f32(
            false, a0, false, b0, (short)0, acc0, false, false);
        acc1 = __builtin_amdgcn_wmma_f32_16x16x4_f32(
            false, a1, false, b1, (short)0, acc1, false, false);
        t += 8;
    }
}

// Logits via WMMA.  One wave handles 16 edges.
//   A (16x4 f32):  lanes 0-15 = edge m, K=0,1 ; lanes 16-31 = edge m, K=2,3
//   B (4x16 f32):  VGPR0 = rows K in {0,2}, VGPR1 = rows K in {1,3}; N=lane&15
//   C (16x16 f32): VGPR r, lanes 0-15 -> M=r, N=lane ; lanes 16-31 -> M=r+8
__global__ void __launch_bounds__(256)
logits_wmma(const float* __restrict__ q, const float* __restrict__ k,
            const float* __restrict__ ewc, float* __restrict__ wbuf, int E) {
    const int lane = threadIdx.x & 31;
    const int wib  = threadIdx.x >> 5;
    const long long tile = (long long)blockIdx.x * (blockDim.x >> 5) + wib;
    const long long base = tile * 16;
    if (base >= E) return;                    // wave-uniform

    const int half = lane >> 4;               // 0: K=0,1   1: K=2,3
    const int m    = lane & 15;               // A row (edge) / B column (head)
    long long em = base + m;                  // tail clamp keeps EXEC full
    if (em >= E) em = E - 1;
    const size_t row = (size_t)em * DFUSE;
    const int    off = 2 * half;

    v8f acc0 = {}, acc1 = {};
    // region 0: cols [  0,128), 32 chunks, span 16
    logits_region<16, 32>(q + row + 0   + off, k + row + 0   + off,
                          off - m * 16, acc0, acc1);
    // region 1: cols [128,320), 48 chunks, span 24
    logits_region<24, 48>(q + row + 128 + off, k + row + 128 + off,
                          off - m * 24, acc0, acc1);
    // region 2: cols [320,480), 40 chunks, span 20
    logits_region<20, 40>(q + row + 320 + off, k + row + 320 + off,
                          off - m * 20, acc0, acc1);

    // C[m'][h]: lane&15 is the head; VGPR r + 8*half is the edge-in-tile
    if (m < NHEADS) {
#pragma unroll
        for (int r = 0; r < 8; ++r) {
            long long e = base + r + half * 8;
            if (e < E) {
                float scale = ewc[e] * INV_SQRT60;
                wbuf[(size_t)e * NHEADS + m] = (acc0[r] + acc1[r]) * scale;
            }
        }
    }
}

// ---------------------------------------------------------------------------
// Per-node aggregation: block = node, thread = fused output column.
// Online softmax per head (running max/sum in registers, redundantly held by
// every thread of the same head).  v reads are fully coalesced 1920B rows.
// Edge list sorted ascending in LDS => deterministic FP accumulation order.
// ---------------------------------------------------------------------------
#define AGG_CAP 2048
__global__ void __launch_bounds__(DFUSE)
aggregate(const float* __restrict__ v, const float* __restrict__ wbuf,
          const int* __restrict__ rowptr, const int* __restrict__ perm,
          float* __restrict__ out) {
    __shared__ int elist[AGG_CAP];
    const int n     = blockIdx.x;
    const int c     = threadIdx.x;            // 0..479
    const int start = rowptr[n];
    const int deg   = rowptr[n + 1] - start;
    const bool fits = (deg <= AGG_CAP);
    const int  m    = fits ? deg : 0;

    for (int j = c; j < m; j += DFUSE) elist[j] = perm[start + j];
    __syncthreads();
    // odd-even transposition sort (deg is small, avg ~20)
    for (int phase = 0; phase < m; ++phase) {
        const int par = phase & 1;
        for (int j = c;; j += DFUSE) {
            int i0 = par + 2 * j;
            if (i0 + 1 >= m) break;
            int a = elist[i0], b = elist[i0 + 1];
            if (a > b) { elist[i0] = b; elist[i0 + 1] = a; }
        }
        __syncthreads();
    }

    const int h = head_of_col(c);
    float mrun = -3.402823466e38f, srun = 0.0f, acc = 0.0f;
    for (int j = 0; j < deg; ++j) {
        const int e = fits ? elist[j] : perm[start + j];
        const float w    = wbuf[(size_t)e * NHEADS + h];
        const float mnew = fmaxf(mrun, w);
        const float f    = __expf(mrun - mnew);   // 0 on first iteration
        const float p    = __expf(w - mnew);
        srun = srun * f + p;
        acc  = acc  * f + p * v[(size_t)e * DFUSE + c];
        mrun = mnew;
    }
    out[(size_t)n * DFUSE + c] = acc / (srun + 1e-16f);
}

// ---------------------------------------------------------------------------
extern "C" void kernel_launch(void* const* d_in, const int* in_sizes, int n_in,
                              void* d_out, int out_size, void* d_ws,
                              size_t ws_size, hipStream_t stream) {
    const float*     key   = (const float*)d_in[0];
    const float*     value = (const float*)d_in[1];
    const float*     query = (const float*)d_in[2];
    const float*     ewc   = (const float*)d_in[3];
    const long long* eidx  = (const long long*)d_in[4];   // [2, E] int64

    const int E = in_sizes[3];          // edge_weight_cutoff has E elements
    const int N = out_size / DFUSE;     // output is [N, 480]
    const long long* dst = eidx + E;    // second row = destinations
    float* out = (float*)d_out;

    // workspace layout (~7.3 MB): wbuf | cnt | rowptr | cursor | perm
    float* wbuf   = (float*)d_ws;                        // E*8 f32
    int*   cnt    = (int*)(wbuf + (size_t)E * NHEADS);   // N
    int*   rowptr = cnt + N;                             // N+1
    int*   cursor = rowptr + N + 1;                      // N
    int*   perm   = cursor + N;                          // E

    zero_i32<<<(3 * N + 1 + 255) / 256, 256, 0, stream>>>(cnt, 3 * N + 1);
    count_deg<<<(E + 255) / 256, 256, 0, stream>>>(dst, cnt, E);
    scan_rowptr<<<1, 1024, 0, stream>>>(cnt, rowptr, N);
    scatter_perm<<<(E + 255) / 256, 256, 0, stream>>>(dst, rowptr, cursor,
                                                      perm, E);

    const int tiles  = (E + 15) / 16;      // 16 edges per wave
    const int blocks = (tiles + 7) / 8;    // 8 waves per 256-thread block
    logits_wmma<<<blocks, 256, 0, stream>>>(query, key, ewc, wbuf, E);

    aggregate<<<N, DFUSE, 0, stream>>>(value, wbuf, rowptr, perm, out);
}